// torchNet_80324478369804
// MI455X (gfx1250) — compile-verified
//
#include <hip/hip_runtime.h>

// ---------------------------------------------------------------------------
// GCN forward (2x GCNConv + mean-pool + linear) for MI455X / gfx1250.
//  * Dense GEMMs: full-fp32 WMMA (V_WMMA_F32_16X16X4_F32, wave32); X tile
//    staged into LDS via the Tensor Data Mover (tensor_load_to_lds + 
//    s_wait_tensorcnt) when the toolchain exposes the builtin.
//  * Sparse parts (degree, edge scatter, pool): float atomics. Working set
//    (~21 MB) fits the 192 MB L2, so the scatter is L2-bandwidth bound and
//    dominates runtime; GEMM FLOPs (~0.66 GFLOP) are negligible.
// ---------------------------------------------------------------------------

typedef __attribute__((ext_vector_type(2))) float v2f;
typedef __attribute__((ext_vector_type(8))) float v8f;
typedef __attribute__((ext_vector_type(4))) unsigned int u32x4;
typedef __attribute__((ext_vector_type(4))) int i32x4;
typedef __attribute__((ext_vector_type(8))) int i32x8;

#define N_NODES   10000
#define N_EDGES   640000
#define N_GRAPHS  64
#define FEAT      128
#define NUM_CLASS 10

// ---------------------------------------------------------------- degree ---
__global__ __launch_bounds__(256) void k_degree(const int* __restrict__ dst,
                                                float* __restrict__ deg, int E) {
    int e = blockIdx.x * 256 + threadIdx.x;
    if (e < E) atomicAdd(&deg[dst[e]], 1.0f);
}

__global__ __launch_bounds__(256) void k_dinv(const float* __restrict__ deg,
                                              float* __restrict__ dinv, int N) {
    int i = blockIdx.x * 256 + threadIdx.x;
    if (i < N) dinv[i] = rsqrtf(deg[i] + 1.0f);   // +1 for the self-loop
}

// ------------------------------------------------------------ WMMA GEMM ----
// H[M x 128] = X[M x 128] @ W[128 x 128], fp32, V_WMMA_F32_16X16X4_F32.
// One block = 256 threads = 8 waves. Block owns a 16-row tile of X staged
// in LDS (8 KB, dynamic so its LDS offset is 0 for the TDM descriptor);
// wave w computes N-tile w (columns w*16..w*16+15).
// Fragment layouts per CDNA5 ISA 7.12.2:
//   A 16x4 f32 : lane L -> row M=L&15 ; VGPR0 = K=khi, VGPR1 = K=khi+1,
//                khi = 2*(L>=16)
//   B 4x16 f32 : lane L -> col N=L&15 ; VGPR0 = K=khi row, VGPR1 = K=khi+1
//   C/D 16x16  : lane L -> col N=L&15 ; VGPR v = row v + 8*(L>=16)
__global__ __launch_bounds__(256) void k_gemm_wmma(const float* __restrict__ X,
                                                   const float* __restrict__ W,
                                                   float* __restrict__ H) {
    extern __shared__ float As[];            // 16*FEAT floats at LDS offset 0
    const int row0 = blockIdx.x * 16;

#if __has_builtin(__builtin_amdgcn_tensor_load_to_lds) && \
    __has_builtin(__builtin_amdgcn_s_wait_tensorcnt)
    // --- TDM staging: wave 0 issues one 2-D tile DMA (16 rows x 128 f32),
    //     waits on TENSORcnt, then the workgroup barrier publishes the tile.
    if (threadIdx.x < 32) {
        const unsigned long long ga =
            (unsigned long long)(const void*)(X + (size_t)row0 * FEAT);
        u32x4 g0;
        g0[0] = 1u;                                    // count=1 valid descriptor
        g0[1] = 0u;                                    // lds_addr = 0 (dynamic LDS)
        g0[2] = (unsigned int)ga;                      // global_addr[31:0]
        g0[3] = (unsigned int)((ga >> 32) & 0x1FFFFFFull)
              | (2u << 30);                            // addr[56:32] | type=2
        i32x8 g1;
        g1[0] = 0x00020000;                            // data_size=4B, wg_mask=0
        g1[1] = (int)(FEAT << 16);                     // tensor_dim0 = 128
        g1[2] = (int)(16 << 16);                       // tensor_dim1 = 16
        g1[3] = (int)(FEAT << 16);                     // tile_dim0   = 128
        g1[4] = 16;                                    // tile_dim1   = 16
        g1[5] = FEAT;                                  // tensor_dim0_stride = 128
        g1[6] = 0;
        g1[7] = 0;
        i32x4 z4 = {0, 0, 0, 0};
#if defined(__clang_major__) && (__clang_major__ >= 23)
        i32x8 z8 = {0, 0, 0, 0, 0, 0, 0, 0};
        __builtin_amdgcn_tensor_load_to_lds(g0, g1, z4, z4, z8, 0);
#else
        __builtin_amdgcn_tensor_load_to_lds(g0, g1, z4, z4, 0);
#endif
        __builtin_amdgcn_s_wait_tensorcnt(0);
    }
#else
    // --- fallback: cooperative vector-load staging
    for (int i = threadIdx.x; i < 16 * FEAT; i += 256)
        As[i] = X[row0 * FEAT + i];
#endif
    __syncthreads();

    const int lane = threadIdx.x & 31;
    const int wid  = threadIdx.x >> 5;          // N-tile 0..7
    const int col  = wid * 16 + (lane & 15);
    const int khi  = (lane >> 4) << 1;          // 0 (lanes 0-15) or 2 (16-31)
    const int arow = lane & 15;

    v8f acc = {};
    for (int k0 = 0; k0 < FEAT; k0 += 4) {
        v2f a, b;
        a.x = As[arow * FEAT + k0 + khi];
        a.y = As[arow * FEAT + k0 + khi + 1];
        b.x = W[(k0 + khi) * FEAT + col];
        b.y = W[(k0 + khi + 1) * FEAT + col];
        acc = __builtin_amdgcn_wmma_f32_16x16x4_f32(
            /*neg_a=*/false, a, /*neg_b=*/false, b,
            /*c_mod=*/(short)0, acc, /*reuse_a=*/false, /*reuse_b=*/false);
    }

    const int rbase = row0 + ((lane >> 4) << 3);
#pragma unroll
    for (int v = 0; v < 8; ++v)
        H[(rbase + v) * FEAT + col] = acc[v];
}

// ------------------------------------------------------------- scatter -----
// One wave per edge: lane l carries features 4l..4l+3 (float4 gather of
// h[src], 4 atomic f32 adds into agg[dst]). norm = dinv[src]*dinv[dst].
__global__ __launch_bounds__(256) void k_scatter(const float* __restrict__ h,
                                                 const int* __restrict__ src,
                                                 const int* __restrict__ dst,
                                                 const float* __restrict__ dinv,
                                                 float* __restrict__ agg, int E) {
    const int lane = threadIdx.x & 31;
    const int e = blockIdx.x * 8 + (threadIdx.x >> 5);
    if (e >= E) return;
    const int s = src[e];
    const int d = dst[e];
    const float scale = dinv[s] * dinv[d];
    const float4 v = *(const float4*)(h + s * FEAT + lane * 4);
    float* o = agg + d * FEAT + lane * 4;
    atomicAdd(o + 0, v.x * scale);
    atomicAdd(o + 1, v.y * scale);
    atomicAdd(o + 2, v.z * scale);
    atomicAdd(o + 3, v.w * scale);
}

// ------------------------------------------------- self-loop + bias + relu -
__global__ __launch_bounds__(256) void k_finalize(const float* __restrict__ h,
                                                  float* __restrict__ agg,
                                                  const float* __restrict__ dinv,
                                                  const float* __restrict__ bias,
                                                  int total) {
    int idx = blockIdx.x * 256 + threadIdx.x;
    if (idx >= total) return;
    int i = idx >> 7;            // /FEAT
    int f = idx & (FEAT - 1);
    float di = dinv[i];
    float v = agg[idx] + h[idx] * di * di + bias[f];
    agg[idx] = v > 0.0f ? v : 0.0f;
}

// ---------------------------------------------------------------- pooling --
__global__ __launch_bounds__(256) void k_pool(const float* __restrict__ a,
                                              const int* __restrict__ batch,
                                              float* __restrict__ psum,
                                              float* __restrict__ pcnt, int N) {
    const int lane = threadIdx.x & 31;
    const int i = blockIdx.x * 8 + (threadIdx.x >> 5);
    if (i >= N) return;
    const int g = batch[i];
    const float4 v = *(const float4*)(a + i * FEAT + lane * 4);
    float* o = psum + g * FEAT + lane * 4;
    atomicAdd(o + 0, v.x);
    atomicAdd(o + 1, v.y);
    atomicAdd(o + 2, v.z);
    atomicAdd(o + 3, v.w);
    if (lane == 0) atomicAdd(&pcnt[g], 1.0f);
}

// ------------------------------------------------------------------ head ---
__global__ __launch_bounds__(256) void k_head(const float* __restrict__ psum,
                                              const float* __restrict__ pcnt,
                                              const float* __restrict__ Wlin,
                                              const float* __restrict__ blin,
                                              float* __restrict__ out) {
    int t = blockIdx.x * 256 + threadIdx.x;
    if (t >= N_GRAPHS * NUM_CLASS) return;
    int g = t / NUM_CLASS;
    int c = t % NUM_CLASS;
    float inv = 1.0f / fmaxf(pcnt[g], 1.0f);
    float s = 0.0f;
    for (int k = 0; k < FEAT; ++k)
        s += psum[g * FEAT + k] * inv * Wlin[k * NUM_CLASS + c];
    out[t] = s + blin[c];
}

// ---------------------------------------------------------------------------
extern "C" void kernel_launch(void* const* d_in, const int* in_sizes, int n_in,
                              void* d_out, int out_size, void* d_ws, size_t ws_size,
                              hipStream_t stream) {
    (void)in_sizes; (void)n_in; (void)out_size; (void)ws_size;

    const float* x     = (const float*)d_in[0];
    const int*   ei    = (const int*)d_in[1];     // [2, E]
    const int*   batch = (const int*)d_in[2];
    const float* W1    = (const float*)d_in[3];
    const float* b1    = (const float*)d_in[4];
    const float* W2    = (const float*)d_in[5];
    const float* b2    = (const float*)d_in[6];
    const float* Wlin  = (const float*)d_in[7];
    const float* blin  = (const float*)d_in[8];
    float*       out   = (float*)d_out;

    const int* src = ei;            // edge_index[0]
    const int* dst = ei + N_EDGES;  // edge_index[1]

    // ---- carve scratch (~20.6 MB) ----
    float* ws   = (float*)d_ws;
    float* h1   = ws;  ws += (size_t)N_NODES * FEAT;
    float* a1   = ws;  ws += (size_t)N_NODES * FEAT;
    float* h2   = ws;  ws += (size_t)N_NODES * FEAT;
    float* a2   = ws;  ws += (size_t)N_NODES * FEAT;
    float* deg  = ws;  ws += N_NODES;
    float* dinv = ws;  ws += N_NODES;
    float* psum = ws;  ws += N_GRAPHS * FEAT;
    float* pcnt = ws;  ws += N_GRAPHS;

    // Re-zero accumulators every call (graph-capture-safe memset nodes).
    hipMemsetAsync(a1,   0, (size_t)N_NODES * FEAT * sizeof(float), stream);
    hipMemsetAsync(a2,   0, (size_t)N_NODES * FEAT * sizeof(float), stream);
    hipMemsetAsync(deg,  0, (size_t)N_NODES * sizeof(float), stream);
    hipMemsetAsync(psum, 0, (size_t)(N_GRAPHS * FEAT + N_GRAPHS) * sizeof(float),
                   stream);  // psum and pcnt are contiguous

    const size_t lds_bytes = 16 * FEAT * sizeof(float);   // 8 KB X tile

    // ---- degree / normalization ----
    k_degree<<<(N_EDGES + 255) / 256, 256, 0, stream>>>(dst, deg, N_EDGES);
    k_dinv<<<(N_NODES + 255) / 256, 256, 0, stream>>>(deg, dinv, N_NODES);

    // ---- layer 1 ----
    k_gemm_wmma<<<N_NODES / 16, 256, lds_bytes, stream>>>(x, W1, h1);
    k_scatter<<<N_EDGES / 8, 256, 0, stream>>>(h1, src, dst, dinv, a1, N_EDGES);
    k_finalize<<<(N_NODES * FEAT + 255) / 256, 256, 0, stream>>>(h1, a1, dinv, b1,
                                                                 N_NODES * FEAT);

    // ---- layer 2 ----
    k_gemm_wmma<<<N_NODES / 16, 256, lds_bytes, stream>>>(a1, W2, h2);
    k_scatter<<<N_EDGES / 8, 256, 0, stream>>>(h2, src, dst, dinv, a2, N_EDGES);
    k_finalize<<<(N_NODES * FEAT + 255) / 256, 256, 0, stream>>>(h2, a2, dinv, b2,
                                                                 N_NODES * FEAT);

    // ---- mean pool + linear head ----
    k_pool<<<N_NODES / 8, 256, 0, stream>>>(a2, batch, psum, pcnt, N_NODES);
    k_head<<<(N_GRAPHS * NUM_CLASS + 255) / 256, 256, 0, stream>>>(psum, pcnt,
                                                                   Wlin, blin, out);
}